// SubtokenEmbedding_67413806678448
// MI455X (gfx1250) — compile-verified
//
#include <hip/hip_runtime.h>
#include <hip/hip_bf16.h>

typedef float v2f __attribute__((ext_vector_type(2)));
typedef float v4f __attribute__((ext_vector_type(4)));
typedef float v8f __attribute__((ext_vector_type(8)));

#define LRDIM 256
#define NPOS 16
#define DDIM 1024
#define NTOK 16384   // B*S = 4*4096

// -------------------------------------------------------------------------
// Kernel 1: T[p][id][j] = sum_i byte_table[id][i] * pos_emb[p][i][j]
// One 256x256x256 f32 GEMM per position p, via v_wmma_f32_16x16x4_f32.
// grid = (16 M-tiles, 16 positions); block = 128 threads = 4 waves;
// each wave computes 4 adjacent 16x16 N-tiles.
// -------------------------------------------------------------------------
__global__ __launch_bounds__(128)
void precompute_T(const float* __restrict__ byte_table,
                  const float* __restrict__ pos_emb,
                  float* __restrict__ T) {
  const int p    = blockIdx.y;
  const int m0   = blockIdx.x * 16;
  const int wave = threadIdx.x >> 5;
  const int lane = threadIdx.x & 31;
  const int half = lane >> 4;      // 0: K 0..1 / M r ; 1: K 2..3 / M r+8
  const int l16  = lane & 15;
  const int koff = half * 2;
  const int n0   = wave * 64;

  v8f acc[4];
#pragma unroll
  for (int t = 0; t < 4; ++t) acc[t] = (v8f){0.f,0.f,0.f,0.f,0.f,0.f,0.f,0.f};

  const float* bt_row = byte_table + (size_t)(m0 + l16) * LRDIM;
  const float* pe_base = pos_emb + (size_t)p * LRDIM * LRDIM;

  for (int k = 0; k < LRDIM; k += 4) {
    v2f a = *(const v2f*)(bt_row + k + koff);       // A[m][k+koff .. +1]
    const float* pe = pe_base + (size_t)(k + koff) * LRDIM;
#pragma unroll
    for (int t = 0; t < 4; ++t) {
      const int n = n0 + t * 16 + l16;
      v2f b;
      b.x = pe[n];                                   // B[k+koff  ][n]
      b.y = pe[LRDIM + n];                           // B[k+koff+1][n]
      acc[t] = __builtin_amdgcn_wmma_f32_16x16x4_f32(
          false, a, false, b, (short)0, acc[t], false, false);
    }
  }

  float* Tp = T + ((size_t)p * 256 + m0) * LRDIM;
#pragma unroll
  for (int t = 0; t < 4; ++t) {
    const int n = n0 + t * 16 + l16;
#pragma unroll
    for (int r = 0; r < 8; ++r) {
      Tp[(size_t)(r + half * 8) * LRDIM + n] = acc[t][r];
    }
  }
}

// -------------------------------------------------------------------------
// Kernel 2: emb[token][:] = sum_p (id>=0 ? T[p][id][:] : 0)
// One wave per token; each lane owns 8 consecutive columns (2x float4).
// grid = 2048 blocks of 256 threads (8 tokens per block).
// -------------------------------------------------------------------------
__global__ __launch_bounds__(256)
void gather_emb(const int* __restrict__ ids,
                const float* __restrict__ T,
                float* __restrict__ emb) {
  const int token = blockIdx.x * 8 + (threadIdx.x >> 5);
  const int lane  = threadIdx.x & 31;

  v4f a0 = (v4f){0.f,0.f,0.f,0.f};
  v4f a1 = (v4f){0.f,0.f,0.f,0.f};

  const int* idt = ids + (size_t)token * NPOS;
#pragma unroll
  for (int p = 0; p < NPOS; ++p) {
    const int id = idt[p];
    if (id >= 0) {
      const float* row = T + ((size_t)(p * 256 + id)) * LRDIM + lane * 8;
      a0 += *(const v4f*)(row);
      a1 += *(const v4f*)(row + 4);
    }
  }
  float* e = emb + (size_t)token * LRDIM + lane * 8;
  *(v4f*)(e)     = a0;
  *(v4f*)(e + 4) = a1;
}

// -------------------------------------------------------------------------
// Kernel 3: x = emb @ W_up^T + b_up, then LayerNorm over D=1024.
// 16 tokens per block; 8 waves, wave w covers N = [w*128, w*128+128)
// as 8 WMMA 16x16 tiles with a 64-step K loop (f32 WMMA, K=4/step).
// Result staged to LDS (with bias), then two-pass LN over the full row.
// -------------------------------------------------------------------------
__global__ __launch_bounds__(256)
void gemm_bias_ln(const float* __restrict__ emb,
                  const float* __restrict__ W_up,
                  const float* __restrict__ b_up,
                  const float* __restrict__ gamma,
                  const float* __restrict__ beta,
                  float* __restrict__ out) {
  __shared__ float xsh[16][DDIM + 4];   // +4 pad: kills 2-way bank conflicts
  __shared__ float red[16][17];
  __shared__ float mu_sh[16];
  __shared__ float rs_sh[16];

  const int tok0 = blockIdx.x * 16;
  const int wave = threadIdx.x >> 5;
  const int lane = threadIdx.x & 31;
  const int half = lane >> 4;
  const int l16  = lane & 15;
  const int koff = half * 2;
  const int n0   = wave * 128;

  v8f acc[8];
#pragma unroll
  for (int t = 0; t < 8; ++t) acc[t] = (v8f){0.f,0.f,0.f,0.f,0.f,0.f,0.f,0.f};

  const float* arow = emb + (size_t)(tok0 + l16) * LRDIM + koff;
  for (int k = 0; k < LRDIM; k += 4) {
    v2f a = *(const v2f*)(arow + k);                 // A[m][k+koff .. +1]
#pragma unroll
    for (int t = 0; t < 8; ++t) {
      const int n = n0 + t * 16 + l16;
      // B[k][n] = W_up[n][k]; W_up row-major [D, LR] -> contiguous pair.
      v2f b = *(const v2f*)(W_up + (size_t)n * LRDIM + k + koff);
      acc[t] = __builtin_amdgcn_wmma_f32_16x16x4_f32(
          false, a, false, b, (short)0, acc[t], false, false);
    }
  }

  // Spill D fragments to LDS with bias added.
#pragma unroll
  for (int t = 0; t < 8; ++t) {
    const int n = n0 + t * 16 + l16;
    const float bias = b_up[n];
#pragma unroll
    for (int r = 0; r < 8; ++r) {
      xsh[r + half * 8][n] = acc[t][r] + bias;
    }
  }
  __syncthreads();

  // LayerNorm: 16 threads per token, two-pass (mean, then variance).
  const int token = threadIdx.x >> 4;
  const int j0    = threadIdx.x & 15;

  float s = 0.f;
  for (int j = j0; j < DDIM; j += 16) s += xsh[token][j];
  red[token][j0] = s;
  __syncthreads();
  if (j0 == 0) {
    float tot = 0.f;
#pragma unroll
    for (int i = 0; i < 16; ++i) tot += red[token][i];
    mu_sh[token] = tot * (1.0f / DDIM);
  }
  __syncthreads();

  const float mu = mu_sh[token];
  float s2 = 0.f;
  for (int j = j0; j < DDIM; j += 16) {
    const float d = xsh[token][j] - mu;
    s2 += d * d;
  }
  red[token][j0] = s2;
  __syncthreads();
  if (j0 == 0) {
    float tot = 0.f;
#pragma unroll
    for (int i = 0; i < 16; ++i) tot += red[token][i];
    rs_sh[token] = rsqrtf(tot * (1.0f / DDIM) + 1e-5f);
  }
  __syncthreads();

  const float rs = rs_sh[token];
  float* orow = out + (size_t)(tok0 + token) * DDIM;
  for (int j = j0; j < DDIM; j += 16) {
    orow[j] = (xsh[token][j] - mu) * rs * gamma[j] + beta[j];
  }
}

// -------------------------------------------------------------------------
extern "C" void kernel_launch(void* const* d_in, const int* in_sizes, int n_in,
                              void* d_out, int out_size, void* d_ws, size_t ws_size,
                              hipStream_t stream) {
  const int*   ids        = (const int*)d_in[0];    // [4,4096,16]
  const float* byte_table = (const float*)d_in[1];  // [256,256]
  const float* pos_emb    = (const float*)d_in[2];  // [16,256,256]
  const float* W_up       = (const float*)d_in[3];  // [1024,256]
  const float* b_up       = (const float*)d_in[4];  // [1024]
  const float* ln_gamma   = (const float*)d_in[5];  // [1024]
  const float* ln_beta    = (const float*)d_in[6];  // [1024]
  float*       out        = (float*)d_out;          // [4,4096,1024]

  float* T   = (float*)d_ws;                        // 16*256*256 f32 = 4 MB
  float* emb = T + (size_t)NPOS * 256 * LRDIM;      // 16384*256 f32 = 16 MB

  dim3 g1(16, 16);
  precompute_T<<<g1, 128, 0, stream>>>(byte_table, pos_emb, T);
  gather_emb<<<NTOK / 8, 256, 0, stream>>>(ids, T, emb);
  gemm_bias_ln<<<NTOK / 16, 256, 0, stream>>>(emb, W_up, b_up,
                                              ln_gamma, ln_beta, out);
}